// QLSTM_65481071395743
// MI455X (gfx1250) — compile-verified
//
#include <hip/hip_runtime.h>

// ---------------------------------------------------------------------------
// QLSTM for MI455X (gfx1250, wave32).
//
// Math simplification (derived from the CNOT-ring permutation):
//   qlayer(ang, phi)[w] = product of m_j = cos(ang_j + phi_j) over parity sets:
//     out0 = m1*m2*m3, out1 = m0*m1, out2 = m0*m1*m2, out3 = m0*m1*m2*m3
//
// Kernel 1: pre[T*B,16] = X[T*B,16] @ Wx_cat^T + bias_cat   (V_WMMA_F32_16X16X4_F32 x4)
//           ~256 MiB of HBM traffic, memory-bound -> minimize per-tile VALU.
// Kernel 2: sequential LSTM recurrence, 1 thread per batch row (8192 chains),
//           reads pre out of L2 (192 MB >= 160 MB working set).
// ---------------------------------------------------------------------------

typedef float v2f __attribute__((ext_vector_type(2)));
typedef float v8f __attribute__((ext_vector_type(8)));

#define T_STEPS 256
#define BATCH   8192
#define DIN     16
#define MROWS   (T_STEPS * BATCH)        // 2,097,152 rows
#define TILES_PER_WAVE 4
#define WAVES_PER_BLOCK 8

// ---------------------------------------------------------------------------
// Kernel 1: X-projection GEMM with f32 WMMA.
//   pre[m, n] = sum_k X[m,k] * Wg[j, k] + bg[j],  n = g*4 + j, g in {f,i,u,o}
// A fragment (16x4 f32): lane L(0-15): M=L, K={0,1}; lane L+16: M=L, K={2,3}
// B fragment (4x16 f32): lane L(0-15): N=L, K={0,1}; lane L+16: N=L, K={2,3}
// C/D (16x16 f32): vgpr r: lanes0-15 -> (M=r, N=lane); lanes16-31 -> (M=r+8)
// ---------------------------------------------------------------------------
__global__ __launch_bounds__(256) void qlstm_xproj_wmma(
    const float* __restrict__ X,
    const float* __restrict__ Wf, const float* __restrict__ bf,
    const float* __restrict__ Wi, const float* __restrict__ bi,
    const float* __restrict__ Wu, const float* __restrict__ bu,
    const float* __restrict__ Wo, const float* __restrict__ bo,
    float* __restrict__ pre)
{
    const int lane = threadIdx.x & 31;
    const int wave = threadIdx.x >> 5;
    const int n    = lane & 15;              // output column (gate-major)
    const int g    = n >> 2;
    const int j    = n & 3;
    const int kk   = (lane >> 4) << 1;       // 0 or 2: K sub-offset in a K=4 chunk

    const float* Wg = (g == 0) ? Wf : (g == 1) ? Wi : (g == 2) ? Wu : Wo;
    const float* bg = (g == 0) ? bf : (g == 1) ? bi : (g == 2) ? bu : bo;
    const float bias = bg[j];

    // B fragments: Wcat[n][k] for the 4 K-chunks (W rows are length 20; cols 0..15 = x part)
    v2f bfrag[4];
#pragma unroll
    for (int kc = 0; kc < 4; ++kc)
        bfrag[kc] = *(const v2f*)(Wg + j * 20 + kc * 4 + kk);

    const int tile0  = (blockIdx.x * WAVES_PER_BLOCK + wave) * TILES_PER_WAVE;
    const int rowHi  = (lane >> 4) << 3;     // 0 for lanes 0-15, 8 for lanes 16-31

    // Per-lane base pointers, advanced by one 16x16 tile (256 floats) per iter.
    // All intra-tile addressing becomes constant byte offsets -> immediate
    // offsets in global_load/store, one 64-bit address pair each.
    const float* aPtr = X   + (size_t)(tile0 * 16 + (lane & 15)) * DIN + kk;
    float*       sPtr = pre + (size_t)(tile0 * 16 + rowHi) * DIN + n;

#pragma unroll 1
    for (int it = 0; it < TILES_PER_WAVE; ++it) {
        v8f acc;
#pragma unroll
        for (int r = 0; r < 8; ++r) acc[r] = bias;   // bias depends on column only

#pragma unroll
        for (int kc = 0; kc < 4; ++kc) {
            v2f a = *(const v2f*)(aPtr + kc * 4);    // byte offsets 0,16,32,48
            acc = __builtin_amdgcn_wmma_f32_16x16x4_f32(
                /*neg_a=*/false, a, /*neg_b=*/false, bfrag[kc],
                /*c_mod=*/(short)0, acc, /*reuse_a=*/false, /*reuse_b=*/false);
        }

#pragma unroll
        for (int r = 0; r < 8; ++r)
            sPtr[r * DIN] = acc[r];                  // byte offsets 0,64,...,448

        aPtr += 16 * DIN;                            // next tile: +1024 B
        sPtr += 16 * DIN;
    }
}

// ---------------------------------------------------------------------------
// Kernel 2: sequential recurrence. One thread per batch row.
// ---------------------------------------------------------------------------
__device__ __forceinline__ float fast_sigmoid(float x) {
    return 1.0f / (1.0f + __expf(-x));
}
__device__ __forceinline__ float fast_tanh(float x) {
    float e = __expf(2.0f * x);
    return (e - 1.0f) / (e + 1.0f);
}

__global__ __launch_bounds__(256) void qlstm_recurrence(
    const float* __restrict__ pre,
    const float* __restrict__ Wf, const float* __restrict__ Wi,
    const float* __restrict__ Wu, const float* __restrict__ Wo,
    const float* __restrict__ phif, const float* __restrict__ phii,
    const float* __restrict__ phiu, const float* __restrict__ phio,
    float* __restrict__ out)
{
    const int b = blockIdx.x * blockDim.x + threadIdx.x;   // 0..8191, exact

    // Recurrent weights Wh[g][w][h] = W_g[w, 16+h] and phi: uniform across
    // threads -> compiler can keep these in scalar regs / broadcast loads.
    const float* Ws[4] = {Wf, Wi, Wu, Wo};
    const float* Ps[4] = {phif, phii, phiu, phio};
    float Wh[4][4][4];
    float ph[4][4];
#pragma unroll
    for (int gg = 0; gg < 4; ++gg) {
#pragma unroll
        for (int w = 0; w < 4; ++w) {
#pragma unroll
            for (int h = 0; h < 4; ++h)
                Wh[gg][w][h] = Ws[gg][w * 20 + 16 + h];
            ph[gg][w] = Ps[gg][w];
        }
    }

    float hx[4] = {0.f, 0.f, 0.f, 0.f};
    float cx[4] = {0.f, 0.f, 0.f, 0.f};

    const float* pPtr = pre + (size_t)b * DIN;     // advances by BATCH*DIN per step
    float*       oPtr = out + (size_t)b * 4;       // advances by BATCH*4  per step

#pragma unroll 1
    for (int t = 0; t < T_STEPS; ++t) {
        float4 a0 = *(const float4*)(pPtr + 0);
        float4 a1 = *(const float4*)(pPtr + 4);
        float4 a2 = *(const float4*)(pPtr + 8);
        float4 a3 = *(const float4*)(pPtr + 12);
        float ang[16] = {a0.x, a0.y, a0.z, a0.w,  a1.x, a1.y, a1.z, a1.w,
                         a2.x, a2.y, a2.z, a2.w,  a3.x, a3.y, a3.z, a3.w};

        float q[4][4];
#pragma unroll
        for (int gg = 0; gg < 4; ++gg) {
            float m[4];
#pragma unroll
            for (int w = 0; w < 4; ++w) {
                float a = ang[gg * 4 + w];
#pragma unroll
                for (int h = 0; h < 4; ++h)
                    a = fmaf(Wh[gg][w][h], hx[h], a);
                m[w] = __cosf(a + ph[gg][w]);
            }
            // qlayer closed form
            q[gg][0] = m[1] * m[2] * m[3];
            q[gg][1] = m[0] * m[1];
            q[gg][2] = q[gg][1] * m[2];
            q[gg][3] = q[gg][2] * m[3];
        }

#pragma unroll
        for (int w = 0; w < 4; ++w) {
            float fg = fast_sigmoid(q[0][w]);
            float ig = fast_sigmoid(q[1][w]);
            float ug = fast_tanh(q[2][w]);
            float og = fast_sigmoid(q[3][w]);
            cx[w] = fg * cx[w] + ig * ug;
            hx[w] = og * fast_tanh(cx[w]);
        }

        *(float4*)oPtr = make_float4(hx[0], hx[1], hx[2], hx[3]);

        pPtr += (size_t)BATCH * DIN;
        oPtr += (size_t)BATCH * 4;
    }

    float* hO = out + (size_t)T_STEPS * BATCH * 4;
    float* cO = hO + (size_t)BATCH * 4;
    *(float4*)(hO + (size_t)b * 4) = make_float4(hx[0], hx[1], hx[2], hx[3]);
    *(float4*)(cO + (size_t)b * 4) = make_float4(cx[0], cx[1], cx[2], cx[3]);
}

// ---------------------------------------------------------------------------
extern "C" void kernel_launch(void* const* d_in, const int* in_sizes, int n_in,
                              void* d_out, int out_size, void* d_ws, size_t ws_size,
                              hipStream_t stream) {
    const float* X    = (const float*)d_in[0];
    const float* Wf   = (const float*)d_in[1];
    const float* bf   = (const float*)d_in[2];
    const float* Wi   = (const float*)d_in[3];
    const float* bi   = (const float*)d_in[4];
    const float* Wu   = (const float*)d_in[5];
    const float* bu   = (const float*)d_in[6];
    const float* Wo   = (const float*)d_in[7];
    const float* bo   = (const float*)d_in[8];
    const float* phif = (const float*)d_in[9];
    const float* phii = (const float*)d_in[10];
    const float* phiu = (const float*)d_in[11];
    const float* phio = (const float*)d_in[12];

    float* pre = (float*)d_ws;   // needs T*B*16 floats = 128 MiB of scratch

    const int tiles  = MROWS / 16;                                    // 131072
    const int blocks = tiles / (WAVES_PER_BLOCK * TILES_PER_WAVE);    // 4096

    qlstm_xproj_wmma<<<dim3(blocks), dim3(256), 0, stream>>>(
        X, Wf, bf, Wi, bi, Wu, bu, Wo, bo, pre);

    qlstm_recurrence<<<dim3(BATCH / 256), dim3(256), 0, stream>>>(
        pre, Wf, Wi, Wu, Wo, phif, phii, phiu, phio, (float*)d_out);
}